// HistogramLoss_10849087390415
// MI455X (gfx1250) — compile-verified
//
#include <hip/hip_runtime.h>
#include <hip/hip_bf16.h>

// HistogramLoss on gfx1250 (MI455X, wave32):
//   S[b,k] = sum_n exp(-(x_n - c_k)^2 / (2*sigma^2))
// computed as D = K2*(x-c)^2 via V_WMMA_F32_16X16X4_F32 with the exp2 scale
// folded into the B matrix:  A = [x^2, x, 1, 0],  B = K2*[1, -2c, c^2, 0]^T,
// so v_exp_f32 consumes WMMA output directly (no VALU scaling).
// Tiny epilogue (hist -> cdf -> mean|dcdf|) in one block.

typedef __attribute__((ext_vector_type(2))) float v2f;
typedef __attribute__((ext_vector_type(8))) float v8f;

#define NPIX    196608             // 3*256*256 pixels per batch image
#define NBATCH  4
#define NBINS   256
#define EPSV    1e-8f
// -1/(2*sigma^2) * log2(e) = -5000 * 1.4426950408889634
#define K2EXP   (-7213.475204444817f)

#define CHUNKS  48                 // gridDim.x
#define WAVES_PER_BLOCK 8          // 256 threads / wave32
#define WAVES_PER_TILE  (CHUNKS * WAVES_PER_BLOCK)   // 384
#define PIX_PER_ITER    (WAVES_PER_TILE * 32)        // 12288, divides NPIX
#define ITERS           (NPIX / PIX_PER_ITER)        // 16 (exact)

__global__ void soft_hist_wmma(const float* __restrict__ pred,
                               const float* __restrict__ tgt,
                               float* __restrict__ S /* [2][4][256] */) {
    __shared__ float sbin[16];

    const int lane  = threadIdx.x & 31;
    const int wave  = threadIdx.x >> 5;
    const int tile  = blockIdx.y;          // 16-bin tile: 0..15
    const int bz    = blockIdx.z;          // 0..7 = batch + 4*input
    const int inp   = bz >> 2;
    const int batch = bz & 3;
    const float* __restrict__ X = (inp ? tgt : pred) + batch * NPIX;

    if (threadIdx.x < 16) sbin[threadIdx.x] = 0.0f;
    __syncthreads();

    // B matrix 4x16 (constant per bin tile), exp2 scale folded in:
    //   VGPR0: lanes 0-15 = row K=0 (K2),      lanes 16-31 = row K=2 (K2*c^2)
    //   VGPR1: lanes 0-15 = row K=1 (-2*K2*c), lanes 16-31 = row K=3 (0)
    const bool  lo = lane < 16;
    const float c  = (float)(tile * 16 + (lane & 15)) * (1.0f / 255.0f);
    v2f bm;
    bm.x = lo ? K2EXP          : K2EXP * c * c;
    bm.y = lo ? -2.0f * K2EXP * c : 0.0f;

    const int waveGlobal = blockIdx.x * WAVES_PER_BLOCK + wave;

    // Each wave consumes 32 pixels/iter as 16 float2's. Lane l and l+16 load
    // the same float2 (duplicate loads hit L0/L2; pixel->tile assignment is
    // irrelevant because we only column-sum). No cross-lane shuffle needed.
    const float2* __restrict__ Xv =
        (const float2*)(X) + waveGlobal * 16 + (lane & 15);

    const v8f zeroC = {};
    float acc0 = 0.0f, acc1 = 0.0f, acc2 = 0.0f, acc3 = 0.0f;

#pragma unroll 2
    for (int it = 0; it < ITERS; ++it) {
        const float2 xv = Xv[it * (PIX_PER_ITER / 2)];

        // A (16x4): lanes 0-15 carry (x^2, x) in K=0,1; lanes 16-31 carry
        // the constant (1, 0) in K=2,3.
        v2f a0, a1;
        a0.x = lo ? xv.x * xv.x : 1.0f;  a0.y = lo ? xv.x : 0.0f;
        a1.x = lo ? xv.y * xv.y : 1.0f;  a1.y = lo ? xv.y : 0.0f;

        // D[m,n] = K2EXP * (x_m - c_n)^2 : 16 pixels x 16 bins / instruction
        v8f d0 = __builtin_amdgcn_wmma_f32_16x16x4_f32(
            false, a0, false, bm, (short)0, zeroC, false, false);
        v8f d1 = __builtin_amdgcn_wmma_f32_16x16x4_f32(
            false, a1, false, bm, (short)0, zeroC, false, false);

        // exp2 straight off the WMMA output; 4 accumulators break the
        // serial add chain behind the TRANS pipe.
        acc0 += __builtin_amdgcn_exp2f(d0[0]);
        acc1 += __builtin_amdgcn_exp2f(d0[1]);
        acc2 += __builtin_amdgcn_exp2f(d0[2]);
        acc3 += __builtin_amdgcn_exp2f(d0[3]);
        acc0 += __builtin_amdgcn_exp2f(d0[4]);
        acc1 += __builtin_amdgcn_exp2f(d0[5]);
        acc2 += __builtin_amdgcn_exp2f(d0[6]);
        acc3 += __builtin_amdgcn_exp2f(d0[7]);
        acc0 += __builtin_amdgcn_exp2f(d1[0]);
        acc1 += __builtin_amdgcn_exp2f(d1[1]);
        acc2 += __builtin_amdgcn_exp2f(d1[2]);
        acc3 += __builtin_amdgcn_exp2f(d1[3]);
        acc0 += __builtin_amdgcn_exp2f(d1[4]);
        acc1 += __builtin_amdgcn_exp2f(d1[5]);
        acc2 += __builtin_amdgcn_exp2f(d1[6]);
        acc3 += __builtin_amdgcn_exp2f(d1[7]);
    }

    float acc = (acc0 + acc1) + (acc2 + acc3);
    // lane l (rows 0-7) + lane l+16 (rows 8-15) cover full column l
    acc += __shfl_xor(acc, 16);
    if (lo) atomicAdd(&sbin[lane], acc);            // ds_add_f32 across 8 waves
    __syncthreads();
    if (threadIdx.x < 16)
        atomicAdd(&S[inp * (NBATCH * NBINS) + batch * NBINS + tile * 16 + threadIdx.x],
                  sbin[threadIdx.x]);               // global_atomic_add_f32
}

__device__ __forceinline__ float block_sum_256(float v, float* sh) {
    const int t = threadIdx.x;
    __syncthreads();
    sh[t] = v;
    __syncthreads();
    for (int off = 128; off > 0; off >>= 1) {
        if (t < off) sh[t] += sh[t + off];
        __syncthreads();
    }
    const float r = sh[0];
    __syncthreads();
    return r;
}

__device__ __forceinline__ float inclusive_scan_256(float v, float* sh) {
    const int t = threadIdx.x;
    __syncthreads();
    sh[t] = v;
    __syncthreads();
    for (int off = 1; off < 256; off <<= 1) {
        const float add = (t >= off) ? sh[t - off] : 0.0f;
        __syncthreads();
        sh[t] += add;
        __syncthreads();
    }
    const float r = sh[t];
    __syncthreads();
    return r;
}

__global__ void hist_finalize(const float* __restrict__ S,
                              float* __restrict__ out) {
    __shared__ float sh[256];
    const int t = threadIdx.x;
    float acc = 0.0f;
    for (int b = 0; b < NBATCH; ++b) {
        const float sp = S[b * NBINS + t];
        const float st = S[NBATCH * NBINS + b * NBINS + t];
        float hp = sp / (sp + EPSV);          // w / (colsum + eps), summed over n
        float ht = st / (st + EPSV);
        const float sumP = block_sum_256(hp, sh);
        const float sumT = block_sum_256(ht, sh);
        hp /= (sumP + EPSV);                  // hist / (hist.sum + eps)
        ht /= (sumT + EPSV);
        const float cp = inclusive_scan_256(hp, sh);   // jnp.cumsum (inclusive)
        const float ct = inclusive_scan_256(ht, sh);
        acc += fabsf(cp - ct);
    }
    const float tot = block_sum_256(acc, sh);
    if (t == 0) out[0] = tot * (1.0f / (NBATCH * NBINS));
}

extern "C" void kernel_launch(void* const* d_in, const int* in_sizes, int n_in,
                              void* d_out, int out_size, void* d_ws, size_t ws_size,
                              hipStream_t stream) {
    const float* pred = (const float*)d_in[0];
    const float* tgt  = (const float*)d_in[1];
    float* S = (float*)d_ws;                  // 2*4*256 floats = 8 KB scratch

    hipMemsetAsync(S, 0, 2 * NBATCH * NBINS * sizeof(float), stream);

    dim3 block(256);                          // 8 waves (wave32)
    dim3 grid(CHUNKS, 16, 8);                 // pixel chunks x bin-tiles x (batch*input)
    soft_hist_wmma<<<grid, block, 0, stream>>>(pred, tgt, S);
    hist_finalize<<<1, 256, 0, stream>>>(S, (float*)d_out);
}